// GraphFeatLearningLayer_15590731284885
// MI455X (gfx1250) — compile-verified
//
#include <hip/hip_runtime.h>

// GraphFeatLearningLayer for MI455X (gfx1250, wave32).
// Kernel 1: X_bar = pc * alpha (output #2) + per-row sum-of-squares (scratch).
// Kernel 2: 32 batched SYRKs G = X_bar * X_bar^T via V_WMMA_F32_16X16X4_F32,
//           fused D/exp/threshold epilogue writing adj (output #1).

typedef __attribute__((ext_vector_type(2))) float v2f;
typedef __attribute__((ext_vector_type(8))) float v8f;

#define BPC   8
#define KW    4
#define NPTS  1024
#define DD    256
#define LDT   36          // padded LDS row stride in floats (conflict-free b64 reads)

// ---------------------------------------------------------------------------
// Kernel 1: one wave (32 lanes) per (b,k,n) row. Each lane handles 8 d-elems.
// ---------------------------------------------------------------------------
__global__ __launch_bounds__(256) void xbar_sq_kernel(
    const float* __restrict__ pc,      // [B, N, D]
    const float* __restrict__ alphas,  // [K, D]
    float* __restrict__ xbar,          // [B, K, N, D]  (lives inside d_out)
    float* __restrict__ sq)            // [B, K, N]     (lives in d_ws)
{
    const int lane = threadIdx.x & 31;
    const int wave = threadIdx.x >> 5;
    const int row  = blockIdx.x * 8 + wave;                 // b*K*N + k*N + n
    const int k    = (row >> 10) & 3;
    const int pcrow = ((row >> 12) << 10) | (row & 1023);   // b*N + n

    const float* src = pc     + (size_t)pcrow * DD + lane * 8;
    const float* al  = alphas + (size_t)k     * DD + lane * 8;
    float*       dst = xbar   + (size_t)row   * DD + lane * 8;

    float4 p0 = *(const float4*)(src);
    float4 p1 = *(const float4*)(src + 4);
    float4 a0 = *(const float4*)(al);
    float4 a1 = *(const float4*)(al + 4);

    float4 x0, x1;
    x0.x = p0.x * a0.x; x0.y = p0.y * a0.y; x0.z = p0.z * a0.z; x0.w = p0.w * a0.w;
    x1.x = p1.x * a1.x; x1.y = p1.y * a1.y; x1.z = p1.z * a1.z; x1.w = p1.w * a1.w;

    *(float4*)(dst)     = x0;
    *(float4*)(dst + 4) = x1;

    float s = x0.x*x0.x + x0.y*x0.y + x0.z*x0.z + x0.w*x0.w
            + x1.x*x1.x + x1.y*x1.y + x1.z*x1.z + x1.w*x1.w;

    // wave32 butterfly reduction
    #pragma unroll
    for (int off = 16; off >= 1; off >>= 1)
        s += __shfl_xor(s, off, 32);

    if (lane == 0) sq[row] = s;
}

// ---------------------------------------------------------------------------
// Kernel 2: 128x128 tile of adj per workgroup (256 thr = 8 waves), one (b,k)
// slice per blockIdx.z. K staged through LDS in chunks of 32.
// Wave w owns a 32x64 sub-tile: 2 m-tiles x 4 n-tiles of 16x16 accumulators.
// ---------------------------------------------------------------------------
__global__ __launch_bounds__(256) void gram_adj_kernel(
    const float* __restrict__ xbar,   // [BK, N, D]
    const float* __restrict__ sq,     // [BK, N]
    const float* __restrict__ sigptr, // scalar sigma
    float* __restrict__ adj)          // [BK, N, N]
{
    __shared__ float As[128 * LDT];
    __shared__ float Bs[128 * LDT];
    __shared__ float sqm[128];
    __shared__ float sqn[128];

    const int bk = blockIdx.z;
    const float* Xb  = xbar + (size_t)bk * NPTS * DD;
    const float* sqb = sq   + (size_t)bk * NPTS;
    float*       out = adj  + (size_t)bk * NPTS * NPTS;

    const int m_base = blockIdx.y * 128;
    const int n_base = blockIdx.x * 128;

    const int t = threadIdx.x;
    if (t < 128) sqm[t]       = sqb[m_base + t];
    else         sqn[t - 128] = sqb[n_base + (t - 128)];

    const int lane = t & 31;
    const int wave = t >> 5;
    const int wm   = (wave & 3) * 32;   // row strip within 128
    const int wn   = (wave >> 2) * 64;  // col strip within 128
    const int lrow = lane & 15;
    const int lk   = (lane >> 4) * 2;   // K sub-offset per A/B fragment layout

    v8f acc[2][4];
    #pragma unroll
    for (int i = 0; i < 2; ++i)
        #pragma unroll
        for (int j = 0; j < 4; ++j)
            #pragma unroll
            for (int e = 0; e < 8; ++e)
                acc[i][j][e] = 0.0f;

    // cooperative tile loading: thread t fills row t/2, 16-float half (t&1)
    const int ldrow = t >> 1;
    const int ldcol = (t & 1) * 16;

    for (int kk = 0; kk < DD; kk += 32) {
        const float* ga = Xb + (size_t)(m_base + ldrow) * DD + kk + ldcol;
        const float* gb = Xb + (size_t)(n_base + ldrow) * DD + kk + ldcol;
        float4 av0 = ((const float4*)ga)[0];
        float4 av1 = ((const float4*)ga)[1];
        float4 av2 = ((const float4*)ga)[2];
        float4 av3 = ((const float4*)ga)[3];
        float4 bv0 = ((const float4*)gb)[0];
        float4 bv1 = ((const float4*)gb)[1];
        float4 bv2 = ((const float4*)gb)[2];
        float4 bv3 = ((const float4*)gb)[3];

        __syncthreads();   // previous chunk's reads done before overwrite
        float* la = &As[ldrow * LDT + ldcol];
        float* lb = &Bs[ldrow * LDT + ldcol];
        ((float4*)la)[0] = av0; ((float4*)la)[1] = av1;
        ((float4*)la)[2] = av2; ((float4*)la)[3] = av3;
        ((float4*)lb)[0] = bv0; ((float4*)lb)[1] = bv1;
        ((float4*)lb)[2] = bv2; ((float4*)lb)[3] = bv3;
        __syncthreads();

        #pragma unroll
        for (int ks = 0; ks < 32; ks += 4) {
            v2f af[2], bf[4];
            #pragma unroll
            for (int i = 0; i < 2; ++i)
                af[i] = *(const v2f*)&As[(wm + i * 16 + lrow) * LDT + ks + lk];
            #pragma unroll
            for (int j = 0; j < 4; ++j)
                bf[j] = *(const v2f*)&Bs[(wn + j * 16 + lrow) * LDT + ks + lk];
            #pragma unroll
            for (int i = 0; i < 2; ++i)
                #pragma unroll
                for (int j = 0; j < 4; ++j)
                    acc[i][j] = __builtin_amdgcn_wmma_f32_16x16x4_f32(
                        /*neg_a=*/false, af[i], /*neg_b=*/false, bf[j],
                        /*c_mod=*/(short)0, acc[i][j],
                        /*reuse_a=*/false, /*reuse_b=*/false);
        }
    }

    // epilogue: D = sq_m + sq_n - 2G; W = exp(-D/sigma); threshold 0.5
    const float invs = 1.0f / sigptr[0];
    #pragma unroll
    for (int i = 0; i < 2; ++i) {
        const int rbase = wm + i * 16 + 8 * (lane >> 4);
        #pragma unroll
        for (int j = 0; j < 4; ++j) {
            const int c  = wn + j * 16 + (lane & 15);
            const float sc = sqn[c];
            #pragma unroll
            for (int e = 0; e < 8; ++e) {
                const int r = rbase + e;
                const float g = acc[i][j][e];
                const float d = sqm[r] + sc - 2.0f * g;
                const float w = __expf(-d * invs);
                out[(size_t)(m_base + r) * NPTS + (n_base + c)] =
                    (w >= 0.5f) ? w : 0.0f;
            }
        }
    }
}

// ---------------------------------------------------------------------------
extern "C" void kernel_launch(void* const* d_in, const int* in_sizes, int n_in,
                              void* d_out, int out_size, void* d_ws, size_t ws_size,
                              hipStream_t stream) {
    const float* pc     = (const float*)d_in[0];  // [8,1024,256]
    const float* alphas = (const float*)d_in[1];  // [4,256]
    const float* sigma  = (const float*)d_in[2];  // scalar

    float* adj  = (float*)d_out;                                   // [8,4,1024,1024]
    float* xbar = adj + (size_t)BPC * KW * NPTS * NPTS;            // [8,4,1024,256]
    float* sq   = (float*)d_ws;                                    // [8,4,1024]

    // Kernel 1: 32768 rows, 8 waves/block -> 4096 blocks
    xbar_sq_kernel<<<dim3(4096), dim3(256), 0, stream>>>(pc, alphas, xbar, sq);

    // Kernel 2: 8x8 tiles of 128x128 per (b,k); 32 (b,k) slices
    gram_adj_kernel<<<dim3(8, 8, BPC * KW), dim3(256), 0, stream>>>(
        xbar, sq, sigma, adj);
}